// OrientedBboxIOU_35416300323279
// MI455X (gfx1250) — compile-verified
//
#include <hip/hip_runtime.h>
#include <stdint.h>
#include <math.h>

#define EPSG 0.01f
#define TWO_PI_F 6.28318530717958647692f

__device__ __forceinline__ float cleanNum(float x) {
  return __builtin_isfinite(x) ? x : 0.0f;
}

__global__ __launch_bounds__(256) void obb_iou_kernel(const float* __restrict__ g1,
                                                      const float* __restrict__ g2,
                                                      float* __restrict__ out, int n) {
  __shared__ alignas(16) float lds[256 * 16];
  const int tid = threadIdx.x;
  const int gid = blockIdx.x * 256 + tid;
  const int idx = (gid < n) ? gid : (n - 1);

  // ---- CDNA5 async global->LDS staging (ASYNCcnt path) ------------------
  // Generic pointers to LDS carry the LDS byte offset in their low 32 bits.
  uint32_t ldsLow = (uint32_t)(uintptr_t)(&lds[0]);
  uint32_t off1 = ldsLow + (uint32_t)tid * 64u;        // this thread's 64B slot
  uint32_t off2 = off1 + 32u;
  uint64_t ga1 = (uint64_t)(uintptr_t)(g1 + (size_t)idx * 8u);
  uint64_t ga2 = (uint64_t)(uintptr_t)(g2 + (size_t)idx * 8u);
  // INST_OFFSET is added to BOTH the LDS and global addresses (ISA 08 §4.4),
  // so one (vdst, vaddr) pair covers each 32B box with offset:0 / offset:16.
  asm volatile("global_load_async_to_lds_b128 %0, %1, off" :: "v"(off1), "v"(ga1) : "memory");
  asm volatile("global_load_async_to_lds_b128 %0, %1, off offset:16" :: "v"(off1), "v"(ga1) : "memory");
  asm volatile("global_load_async_to_lds_b128 %0, %1, off" :: "v"(off2), "v"(ga2) : "memory");
  asm volatile("global_load_async_to_lds_b128 %0, %1, off offset:16" :: "v"(off2), "v"(ga2) : "memory");
  asm volatile("s_wait_asynccnt 0x0" ::: "memory");

  const float* my = &lds[tid * 16];
  float r1u[4], r1v[4], r2u[4], r2v[4];   // comp0 = u, comp1 = v (matches ref indexing)
#pragma unroll
  for (int c = 0; c < 4; ++c) {
    r1u[c] = my[2 * c + 0];
    r1v[c] = my[2 * c + 1];
    r2u[c] = my[8 + 2 * c + 0];
    r2v[c] = my[8 + 2 * c + 1];
  }

  // ---- edge precompute: a = x1*y2 - y1*x2, deltas -----------------------
  float A1[4], DU1[4], DV1[4], A2[4], DU2[4], DV2[4];
#pragma unroll
  for (int i = 0; i < 4; ++i) {
    int ip = (i + 1) & 3;
    A1[i]  = r1u[i] * r1v[ip] - r1v[i] * r1u[ip];
    DU1[i] = r1u[i] - r1u[ip];
    DV1[i] = r1v[i] - r1v[ip];
    A2[i]  = r2u[i] * r2v[ip] - r2v[i] * r2u[ip];
    DU2[i] = r2u[i] - r2u[ip];
    DV2[i] = r2v[i] - r2v[ip];
  }

  // ---- 16 intersection points + 8 corners (k = i*4 + j, then r1, r2) ----
  float pu[24], pv[24];
#pragma unroll
  for (int i = 0; i < 4; ++i) {
#pragma unroll
    for (int j = 0; j < 4; ++j) {
      float den = DU1[i] * DV2[j] - DV1[i] * DU2[j];
      den = (fabsf(den) < 1e-12f) ? 1e-12f : den;
      pu[i * 4 + j] = (A1[i] * DU2[j] - DU1[i] * A2[j]) / den;
      pv[i * 4 + j] = (A1[i] * DV2[j] - DV1[i] * A2[j]) / den;
    }
  }
#pragma unroll
  for (int c = 0; c < 4; ++c) {
    pu[16 + c] = r1u[c]; pv[16 + c] = r1v[c];
    pu[20 + c] = r2u[c]; pv[20 + c] = r2v[c];
  }

  // ---- point-in-rect tests (eps = 0.01) ---------------------------------
  float ABu1 = r1u[1] - r1u[0], ABv1 = r1v[1] - r1v[0];
  float BCu1 = r1u[2] - r1u[1], BCv1 = r1v[2] - r1v[1];
  float ABAB1 = ABu1 * ABu1 + ABv1 * ABv1;
  float BCBC1 = BCu1 * BCu1 + BCv1 * BCv1;
  float ABu2 = r2u[1] - r2u[0], ABv2 = r2v[1] - r2v[0];
  float BCu2 = r2u[2] - r2u[1], BCv2 = r2v[2] - r2v[1];
  float ABAB2 = ABu2 * ABu2 + ABv2 * ABv2;
  float BCBC2 = BCu2 * BCu2 + BCv2 * BCv2;

  auto inR1 = [&](float u, float v) -> bool {
    float d1 = ABu1 * (u - r1u[0]) + ABv1 * (v - r1v[0]);
    float d2 = BCu1 * (u - r1u[1]) + BCv1 * (v - r1v[1]);
    return (0.0f <= d1 + EPSG) && (d1 <= ABAB1 + EPSG) &&
           (0.0f <= d2 + EPSG) && (d2 <= BCBC1 + EPSG);
  };
  auto inR2 = [&](float u, float v) -> bool {
    float d1 = ABu2 * (u - r2u[0]) + ABv2 * (v - r2v[0]);
    float d2 = BCu2 * (u - r2u[1]) + BCv2 * (v - r2v[1]);
    return (0.0f <= d1 + EPSG) && (d1 <= ABAB2 + EPSG) &&
           (0.0f <= d2 + EPSG) && (d2 <= BCBC2 + EPSG);
  };

  bool msk[24];
#pragma unroll
  for (int k = 0; k < 16; ++k)
    msk[k] = inR1(pu[k], pv[k]) && inR2(pu[k], pv[k]);
#pragma unroll
  for (int c = 0; c < 4; ++c) {
    msk[16 + c] = inR2(r1u[c], r1v[c]);   // rect1 corners inside rect2
    msk[20 + c] = inR1(r2u[c], r2v[c]);   // rect2 corners inside rect1
  }

  // ---- duplicate suppression (L1 dist < 0.01 vs any earlier point) ------
  bool mi[24];
#pragma unroll
  for (int k = 0; k < 24; ++k) {
    bool dup = false;
#pragma unroll
    for (int j = 0; j < 24; ++j) {
      if (j < k)
        dup = dup || ((fabsf(pu[k] - pu[j]) + fabsf(pv[k] - pv[j])) < EPSG);
    }
    mi[k] = msk[k] && !dup;
  }

  // ---- stable top-8 of mask (== first 8 mask-1 points, ascending k) -----
  // Filler rows (mask==0) are replaced by origin in convex_polygon_area, so
  // their coordinates never matter; only the count and selected points do.
  float su[8], sv[8];
#pragma unroll
  for (int s = 0; s < 8; ++s) { su[s] = 0.0f; sv[s] = 0.0f; }
  int cnt = 0;
#pragma unroll
  for (int k = 0; k < 24; ++k) {
    float cu = cleanNum(pu[k]);   // NaN/Inf -> 0 (per component, as reference)
    float cv = cleanNum(pv[k]);
    bool take = mi[k] && (cnt < 8);
#pragma unroll
    for (int s = 0; s < 8; ++s) {
      bool w = take && (cnt == s);
      su[s] = w ? cu : su[s];
      sv[s] = w ? cv : sv[s];
    }
    cnt += take ? 1 : 0;
  }

  // ---- convex polygon area ----------------------------------------------
  float ou = su[0], ov = sv[0];               // origin = vertices[0]
  float ppu[8], ppv[8], sumu = 0.0f, sumv = 0.0f;
#pragma unroll
  for (int s = 0; s < 8; ++s) {
    bool on = (s < cnt);
    float u = on ? su[s] : ou;
    float v = on ? sv[s] : ov;
    ppu[s] = u; ppv[s] = v;
    sumu += on ? u : 0.0f;
    sumv += on ? v : 0.0f;
  }
  float inv = 1.0f / ((float)cnt + 1e-6f);
  float cenu = sumu * inv, cenv = sumv * inv;

  float ang[8]; int id[8];
#pragma unroll
  for (int s = 0; s < 8; ++s) {
    ppu[s] -= cenu; ppv[s] -= cenv;
    ang[s] = atan2f(ppu[s], ppv[s]);          // arctan2(comp0, comp1) as reference
    id[s] = s;
  }
  float a0 = ang[0];
#pragma unroll
  for (int s = 0; s < 8; ++s) {
    float a = ang[s] - a0;
    ang[s] = (a < 0.0f) ? a + TWO_PI_F : a;
  }

  // Batcher 8-input sorting network (19 CEs), stable via (angle, index) key.
  auto ce = [&](int i, int j) {
    bool sw = (ang[i] > ang[j]) || ((ang[i] == ang[j]) && (id[i] > id[j]));
    float ta = ang[i], tu = ppu[i], tv = ppv[i]; int ti = id[i];
    ang[i] = sw ? ang[j] : ang[i];
    ppu[i] = sw ? ppu[j] : ppu[i];
    ppv[i] = sw ? ppv[j] : ppv[i];
    id[i]  = sw ? id[j]  : id[i];
    ang[j] = sw ? ta : ang[j];
    ppu[j] = sw ? tu : ppu[j];
    ppv[j] = sw ? tv : ppv[j];
    id[j]  = sw ? ti : id[j];
  };
  ce(0,1); ce(2,3); ce(4,5); ce(6,7);
  ce(0,2); ce(1,3); ce(4,6); ce(5,7);
  ce(1,2); ce(5,6);
  ce(0,4); ce(1,5); ce(2,6); ce(3,7);
  ce(2,4); ce(3,5);
  ce(1,2); ce(3,4); ce(5,6);

  // Fan triangulation: (v0, vi, vi+1), i = 1..6; y = comp0 (u), x = comp1 (v)
  float inter = 0.0f;
#pragma unroll
  for (int i = 1; i <= 6; ++i) {
    float t = ppu[0] * (ppv[i] - ppv[i + 1]) +
              ppu[i] * (ppv[i + 1] - ppv[0]) +
              ppu[i + 1] * (ppv[0] - ppv[i]);
    inter += 0.5f * fabsf(t);
  }

  // ---- rectangle areas & IoU --------------------------------------------
  float areaA = fabsf(r1u[0] * (r1v[1] - r1v[2]) + r1u[1] * (r1v[2] - r1v[0]) +
                      r1u[2] * (r1v[0] - r1v[1]));
  float areaB = fabsf(r2u[0] * (r2v[1] - r2v[2]) + r2u[1] * (r2v[2] - r2v[0]) +
                      r2u[2] * (r2v[0] - r2v[1]));
  float uni = areaA + areaB - inter;

  if (gid < n) out[gid] = inter / uni;
}

extern "C" void kernel_launch(void* const* d_in, const int* in_sizes, int n_in,
                              void* d_out, int out_size, void* d_ws, size_t ws_size,
                              hipStream_t stream) {
  const float* b1 = (const float*)d_in[0];
  const float* b2 = (const float*)d_in[1];
  float* out = (float*)d_out;
  int n = in_sizes[0] / 8;            // (N, 4, 2) float32 per box array
  if (n <= 0) return;
  int blocks = (n + 255) / 256;
  obb_iou_kernel<<<blocks, 256, 0, stream>>>(b1, b2, out, n);
}